// GCN_10273561772520
// MI455X (gfx1250) — compile-verified
//
#include <hip/hip_runtime.h>

#define N_NODES 100000
#define N_EDGES 1600000
#define IN_F    128
#define H_F     64

typedef __attribute__((ext_vector_type(2))) float v2f;
typedef __attribute__((ext_vector_type(8))) float v8f;

// ---------------- utility kernels ----------------

__global__ void gcn_zero_f32(float* __restrict__ p, long n) {
    long i = (long)blockIdx.x * blockDim.x + threadIdx.x;
    if (i < n) p[i] = 0.f;
}

__global__ void gcn_degree_accum(const int* __restrict__ src,
                                 const int* __restrict__ dst,
                                 float* __restrict__ deg_src,
                                 float* __restrict__ deg_dst, int ne) {
    int e = blockIdx.x * blockDim.x + threadIdx.x;
    if (e < ne) {
        atomicAdd(&deg_src[src[e]], 1.0f);
        atomicAdd(&deg_dst[dst[e]], 1.0f);
    }
}

// deg -> 1/sqrt(max(deg,1)) in place (operates over both arrays contiguously)
__global__ void gcn_inv_sqrt(float* __restrict__ deg, long n) {
    long i = (long)blockIdx.x * blockDim.x + threadIdx.x;
    if (i < n) deg[i] = rsqrtf(fmaxf(deg[i], 1.0f));
}

// ---------------- fp32 WMMA GEMM: C[M,64] = (rowscale * A[M,K]) @ W[K,64] ----------------
// Block: 128 threads = 4 waves. blockIdx.x = M-tile (16 rows). Wave w owns N-tile w (16 cols).
// Uses V_WMMA_F32_16X16X4_F32, accumulating over K in steps of 4.

template <int K>
__global__ void gcn_gemm_wmma(const float* __restrict__ A,
                              const float* __restrict__ W,
                              const float* __restrict__ rowscale,
                              float* __restrict__ C) {
    const int wave = threadIdx.x >> 5;      // 0..3 -> N tile
    const int lane = threadIdx.x & 31;
    const int half = lane >> 4;             // 0 or 1
    const int l16  = lane & 15;
    const int m0   = blockIdx.x * 16;
    const int n0   = wave * 16;
    const int m    = m0 + l16;              // A row owned by this lane

    const float s = rowscale[m];            // out-degree norm fused into A
    const float* __restrict__ arow = A + (long)m * K;

    v8f c = {};
    #pragma unroll 4
    for (int k0 = 0; k0 < K; k0 += 4) {
        const int ka = k0 + 2 * half;       // lanes 0-15: K=k0,k0+1 ; lanes 16-31: K=k0+2,k0+3
        v2f a, b;
        a.x = arow[ka]     * s;
        a.y = arow[ka + 1] * s;
        b.x = W[(long)ka       * H_F + n0 + l16];
        b.y = W[(long)(ka + 1) * H_F + n0 + l16];
        // 8 args: (neg_a, A, neg_b, B, c_mod, C, reuse_a, reuse_b)
        c = __builtin_amdgcn_wmma_f32_16x16x4_f32(
                false, a, false, b, (short)0, c, false, false);
    }

    // C/D layout: VGPR r -> row (r + 8*half), col (n0 + l16)
    #pragma unroll
    for (int r = 0; r < 8; ++r) {
        C[(long)(m0 + r + 8 * half) * H_F + n0 + l16] = c[r];
    }
}

// ---------------- SpMM scatter: agg[dst[e]] += h[src[e]]  (64 f32 per edge) ----------------
// One edge per 16 lanes; each lane moves a float4 (coalesced 256B row per edge).

__global__ void gcn_spmm_scatter(const float* __restrict__ h,
                                 const int* __restrict__ src,
                                 const int* __restrict__ dst,
                                 float* __restrict__ agg, int ne) {
    long t = (long)blockIdx.x * blockDim.x + threadIdx.x;
    int e = (int)(t >> 4);
    int c = (int)(t & 15) << 2;
    if (e < ne) {
        const int s = src[e];
        const int d = dst[e];
        const float4 v = *(const float4*)(h + (long)s * H_F + c);
        float* p = agg + (long)d * H_F + c;
        atomicAdd(p + 0, v.x);
        atomicAdd(p + 1, v.y);
        atomicAdd(p + 2, v.z);
        atomicAdd(p + 3, v.w);
    }
}

// ---------------- h = relu(agg * in_norm[node] + bias[feat]) ----------------

__global__ void gcn_bias_relu(const float* __restrict__ agg,
                              const float* __restrict__ in_norm,
                              const float* __restrict__ bias,
                              float* __restrict__ h, long n) {
    long i = (long)blockIdx.x * blockDim.x + threadIdx.x;
    if (i < n) {
        int node = (int)(i >> 6);
        int f = (int)(i & 63);
        float v = agg[i] * in_norm[node] + bias[f];
        h[i] = fmaxf(v, 0.0f);
    }
}

// ---------------- head: out[n,2] = h[n,:] @ Wm[64,2] + bm ----------------

__global__ void gcn_head(const float* __restrict__ h,
                         const float* __restrict__ Wm,
                         const float* __restrict__ bm,
                         float* __restrict__ out, int n) {
    int node = blockIdx.x * blockDim.x + threadIdx.x;
    if (node < n) {
        float a0 = bm[0], a1 = bm[1];
        const float* __restrict__ hr = h + (long)node * H_F;
        #pragma unroll 8
        for (int k = 0; k < H_F; ++k) {
            float v = hr[k];
            a0 = fmaf(v, Wm[k * 2 + 0], a0);
            a1 = fmaf(v, Wm[k * 2 + 1], a1);
        }
        out[node * 2 + 0] = a0;
        out[node * 2 + 1] = a1;
    }
}

// ---------------- launch ----------------

extern "C" void kernel_launch(void* const* d_in, const int* in_sizes, int n_in,
                              void* d_out, int out_size, void* d_ws, size_t ws_size,
                              hipStream_t stream) {
    const float* feature = (const float*)d_in[0];  // [100000,128]
    const float* W1      = (const float*)d_in[1];  // [128,64]
    const float* b1      = (const float*)d_in[2];  // [64]
    const float* W2      = (const float*)d_in[3];  // [64,64]
    const float* b2      = (const float*)d_in[4];  // [64]
    const float* Wm      = (const float*)d_in[5];  // [64,2]
    const float* bm      = (const float*)d_in[6];  // [2]
    const int*   src     = (const int*)d_in[7];    // [1.6M]
    const int*   dst     = (const int*)d_in[8];    // [1.6M]
    float*       out     = (float*)d_out;          // [100000,2]

    float* out_norm = (float*)d_ws;                 // N_NODES
    float* in_norm  = out_norm + N_NODES;           // N_NODES
    float* buf0     = in_norm + N_NODES;            // N_NODES*64
    float* buf1     = buf0 + (long)N_NODES * H_F;   // N_NODES*64

    const long NH = (long)N_NODES * H_F;            // 6.4M
    const int  B  = 256;

    // 1) degrees -> inverse sqrt norms (both arrays are contiguous)
    gcn_zero_f32<<<(2 * N_NODES + B - 1) / B, B, 0, stream>>>(out_norm, 2 * N_NODES);
    gcn_degree_accum<<<(N_EDGES + B - 1) / B, B, 0, stream>>>(src, dst, out_norm, in_norm, N_EDGES);
    gcn_inv_sqrt<<<(2 * N_NODES + B - 1) / B, B, 0, stream>>>(out_norm, 2 * N_NODES);

    // 2) layer 1: GEMM (WMMA) -> scatter -> bias/norm/relu
    gcn_gemm_wmma<IN_F><<<N_NODES / 16, 128, 0, stream>>>(feature, W1, out_norm, buf0);
    gcn_zero_f32<<<(int)((NH + B - 1) / B), B, 0, stream>>>(buf1, NH);
    gcn_spmm_scatter<<<(int)(((long)N_EDGES * 16 + B - 1) / B), B, 0, stream>>>(buf0, src, dst, buf1, N_EDGES);
    gcn_bias_relu<<<(int)((NH + B - 1) / B), B, 0, stream>>>(buf1, in_norm, b1, buf0, NH);

    // 3) layer 2
    gcn_gemm_wmma<H_F><<<N_NODES / 16, 128, 0, stream>>>(buf0, W2, out_norm, buf1);
    gcn_zero_f32<<<(int)((NH + B - 1) / B), B, 0, stream>>>(buf0, NH);
    gcn_spmm_scatter<<<(int)(((long)N_EDGES * 16 + B - 1) / B), B, 0, stream>>>(buf1, src, dst, buf0, N_EDGES);
    gcn_bias_relu<<<(int)((NH + B - 1) / B), B, 0, stream>>>(buf0, in_norm, b2, buf1, NH);

    // 4) head
    gcn_head<<<(N_NODES + B - 1) / B, B, 0, stream>>>(buf1, Wm, bm, out, N_NODES);
}